// GATBlock_33105607917905
// MI455X (gfx1250) — compile-verified
//
#include <hip/hip_runtime.h>
#include <hip/hip_bf16.h>

// ---------------------------------------------------------------------------
// GAT block for MI455X (gfx1250, wave32).
//   h = x @ W                      -> V_WMMA_F32_16X16X4_F32 (native fp32 WMMA)
//   a_src/a_dst per-node logits    -> per-thread dot(32)
//   edge softmax (segment max/sum) -> u32-encoded atomicMax + f32 atomicAdd (L2)
//   message aggregation            -> float4 gather of h[src], f32 atomicAdds
//   epilogue                       -> bias + residual + ELU + LayerNorm
// ---------------------------------------------------------------------------

typedef float v2f __attribute__((ext_vector_type(2)));
typedef float v8f __attribute__((ext_vector_type(8)));

#define IN_DIM 256
#define HEADS 8
#define CPH 32          // channels per head
#define NEG_SLOPE 0.2f
#define LN_EPS 1e-5f

// ---------------- order-preserving float <-> uint encoding (for atomicMax) ---
__device__ __forceinline__ unsigned enc_f32(float f) {
  unsigned b = __float_as_uint(f);
  return (b & 0x80000000u) ? ~b : (b | 0x80000000u);
}
__device__ __forceinline__ float dec_f32(unsigned u) {
  return (u & 0x80000000u) ? __uint_as_float(u ^ 0x80000000u)
                           : __uint_as_float(~u);
}

// ---------------- zero fill --------------------------------------------------
__global__ void fill_zero_u32(unsigned* __restrict__ p, long long n) {
  long long i = (long long)blockIdx.x * blockDim.x + threadIdx.x;
  long long stride = (long long)gridDim.x * blockDim.x;
  for (; i < n; i += stride) p[i] = 0u;
}

// ---------------- GEMM: h = x @ W  via V_WMMA_F32_16X16X4_F32 ----------------
// Block = 128 threads = 4 waves. Wave w computes rows [blk*64 + w*16, +16),
// cols [blockIdx.y*64, +64) as four 16x16 tiles (4 independent accumulators).
__global__ __launch_bounds__(128)
void gemm_xw_wmma(const float* __restrict__ x, const float* __restrict__ W,
                  float* __restrict__ h, int nrows) {
  const int lane = threadIdx.x & 31;
  const int wave = threadIdx.x >> 5;
  const int lm = lane & 15;   // M (for A) / N (for B,D) index within tile
  const int lg = lane >> 4;   // half-wave selector
  const int rowBase = blockIdx.x * 64 + wave * 16;
  const int colBase = blockIdx.y * 64;

  // A-fragment source row for this lane (clamped; stores are guarded)
  int arow = rowBase + lm;
  if (arow >= nrows) arow = nrows - 1;
  const float* __restrict__ xrow = x + (size_t)arow * IN_DIM;

  v8f acc0 = {}; v8f acc1 = {}; v8f acc2 = {}; v8f acc3 = {};

#pragma unroll 4
  for (int k = 0; k < IN_DIM; k += 4) {
    const int ak = k + 2 * lg;                       // this lane's K pair
    v2f a = *(const v2f*)(xrow + ak);                // A: K=ak, ak+1 (8B aligned)

    const float* __restrict__ wr0 = W + (size_t)ak * IN_DIM;        // row ak
    const float* __restrict__ wr1 = W + (size_t)(ak + 1) * IN_DIM;  // row ak+1
    const int c0 = colBase + lm;
    v2f b0 = { wr0[c0 +  0], wr1[c0 +  0] };
    v2f b1 = { wr0[c0 + 16], wr1[c0 + 16] };
    v2f b2 = { wr0[c0 + 32], wr1[c0 + 32] };
    v2f b3 = { wr0[c0 + 48], wr1[c0 + 48] };

    acc0 = __builtin_amdgcn_wmma_f32_16x16x4_f32(false, a, false, b0, (short)0, acc0, false, false);
    acc1 = __builtin_amdgcn_wmma_f32_16x16x4_f32(false, a, false, b1, (short)0, acc1, false, false);
    acc2 = __builtin_amdgcn_wmma_f32_16x16x4_f32(false, a, false, b2, (short)0, acc2, false, false);
    acc3 = __builtin_amdgcn_wmma_f32_16x16x4_f32(false, a, false, b3, (short)0, acc3, false, false);
  }

  // D layout: lane holds N=lm, rows M = v + 8*lg for v = 0..7
  const int srow = rowBase + 8 * lg;
#pragma unroll
  for (int v = 0; v < 8; ++v) {
    const int r = srow + v;
    if (r < nrows) {
      float* __restrict__ hp = h + (size_t)r * IN_DIM + colBase + lm;
      hp[ 0] = acc0[v];
      hp[16] = acc1[v];
      hp[32] = acc2[v];
      hp[48] = acc3[v];
    }
  }
}

// ---------------- per-node attention logits ---------------------------------
__global__ void node_logits(const float* __restrict__ h,
                            const float* __restrict__ att_src,
                            const float* __restrict__ att_dst,
                            float* __restrict__ a_src, float* __restrict__ a_dst,
                            int n) {
  int t = blockIdx.x * blockDim.x + threadIdx.x;   // t = node*8 + head
  if (t >= n * HEADS) return;
  const int node = t >> 3, hd = t & 7;
  const float4* hp = (const float4*)(h + (size_t)node * IN_DIM + hd * CPH);
  const float4* as = (const float4*)(att_src + hd * CPH);
  const float4* ad = (const float4*)(att_dst + hd * CPH);
  float s = 0.f, d = 0.f;
#pragma unroll
  for (int c = 0; c < CPH / 4; ++c) {
    float4 v = hp[c], a = as[c], b = ad[c];
    s += v.x * a.x + v.y * a.y + v.z * a.z + v.w * a.w;
    d += v.x * b.x + v.y * b.y + v.z * b.z + v.w * b.w;
  }
  a_src[t] = s;
  a_dst[t] = d;
}

// ---------------- edge logits + segment max ---------------------------------
__global__ void edge_alpha_max(const long long* __restrict__ ei,
                               const float* __restrict__ a_src,
                               const float* __restrict__ a_dst,
                               float* __restrict__ alpha_buf,     // [E*8]
                               unsigned* __restrict__ amax_enc,   // [N*8]
                               long long E) {
  long long t = (long long)blockIdx.x * blockDim.x + threadIdx.x;
  if (t >= E * HEADS) return;
  const long long e = t >> 3;
  const int hd = (int)(t & 7);
  const long long s = ei[e], d = ei[E + e];
  float al = a_src[s * HEADS + hd] + a_dst[d * HEADS + hd];
  al = (al > 0.f) ? al : NEG_SLOPE * al;            // LeakyReLU
  alpha_buf[t] = al;
  atomicMax(&amax_enc[d * HEADS + hd], enc_f32(al));
}

// ---------------- exp(alpha - max) + segment sum -----------------------------
__global__ void edge_exp_sum(const long long* __restrict__ ei,
                             float* __restrict__ alpha_buf,       // in: alpha, out: ex
                             const unsigned* __restrict__ amax_enc,
                             float* __restrict__ denom,           // [N*8]
                             long long E) {
  long long t = (long long)blockIdx.x * blockDim.x + threadIdx.x;
  if (t >= E * HEADS) return;
  const long long e = t >> 3;
  const int hd = (int)(t & 7);
  const long long d = ei[E + e];
  const float m = dec_f32(amax_enc[d * HEADS + hd]);
  const float ex = __expf(alpha_buf[t] - m);
  alpha_buf[t] = ex;
  atomicAdd(&denom[d * HEADS + hd], ex);
}

// ---------------- weighted message scatter -----------------------------------
// One thread handles 4 channels of one edge (all within one head: 4 | 32).
__global__ void edge_scatter(const long long* __restrict__ ei,
                             const float* __restrict__ h,
                             const float* __restrict__ exbuf,     // [E*8]
                             const float* __restrict__ denom,     // [N*8]
                             float* __restrict__ out,             // [N*256] accum
                             long long E) {
  const long long total = E * (IN_DIM / 4);
  long long t = (long long)blockIdx.x * blockDim.x + threadIdx.x;
  const long long stride = (long long)gridDim.x * blockDim.x;
  for (; t < total; t += stride) {
    const long long e = t >> 6;            // 64 quad-threads per edge
    const int q = (int)(t & 63);
    const int c = q * 4;
    const int hd = c >> 5;                 // head index
    const long long s = ei[e], d = ei[E + e];
    const float attn = exbuf[e * HEADS + hd] / (denom[d * HEADS + hd] + 1e-16f);
    const float4 hv = *(const float4*)(h + s * IN_DIM + c);
    float* __restrict__ op = out + d * IN_DIM + c;
    atomicAdd(op + 0, hv.x * attn);
    atomicAdd(op + 1, hv.y * attn);
    atomicAdd(op + 2, hv.z * attn);
    atomicAdd(op + 3, hv.w * attn);
  }
}

// ---------------- epilogue: bias + residual + ELU + LayerNorm ----------------
__global__ __launch_bounds__(256)
void epilogue_ln(const float* __restrict__ x, const float* __restrict__ bias,
                 const float* __restrict__ ln_scale, const float* __restrict__ ln_bias,
                 float* __restrict__ out, int n) {
  const int node = blockIdx.x;
  if (node >= n) return;
  const int c = threadIdx.x;
  float v = out[(size_t)node * IN_DIM + c] + bias[c] + x[(size_t)node * IN_DIM + c];
  v = (v > 0.f) ? v : (__expf(v) - 1.0f);           // ELU

  // block reduction: 8 wave32 partials -> LDS -> final
  float sum = v, sq = v * v;
#pragma unroll
  for (int off = 16; off > 0; off >>= 1) {
    sum += __shfl_xor(sum, off);
    sq  += __shfl_xor(sq,  off);
  }
  __shared__ float ssum[8], ssq[8];
  const int wv = threadIdx.x >> 5, ln = threadIdx.x & 31;
  if (ln == 0) { ssum[wv] = sum; ssq[wv] = sq; }
  __syncthreads();
  if (wv == 0) {
    float a = (ln < 8) ? ssum[ln] : 0.f;
    float b = (ln < 8) ? ssq[ln]  : 0.f;
#pragma unroll
    for (int off = 4; off > 0; off >>= 1) {
      a += __shfl_xor(a, off);
      b += __shfl_xor(b, off);
    }
    if (ln == 0) { ssum[0] = a; ssq[0] = b; }
  }
  __syncthreads();
  const float mu  = ssum[0] * (1.0f / IN_DIM);
  const float var = ssq[0]  * (1.0f / IN_DIM) - mu * mu;
  const float rs  = rsqrtf(var + LN_EPS);
  out[(size_t)node * IN_DIM + c] = (v - mu) * rs * ln_scale[c] + ln_bias[c];
}

// ---------------------------------------------------------------------------
extern "C" void kernel_launch(void* const* d_in, const int* in_sizes, int n_in,
                              void* d_out, int out_size, void* d_ws, size_t ws_size,
                              hipStream_t stream) {
  const float*      x        = (const float*)d_in[0];
  const long long*  ei       = (const long long*)d_in[1];   // int64 [2, E]
  const float*      W        = (const float*)d_in[2];
  const float*      att_src  = (const float*)d_in[3];
  const float*      att_dst  = (const float*)d_in[4];
  const float*      bias     = (const float*)d_in[5];
  const float*      ln_scale = (const float*)d_in[6];
  const float*      ln_bias  = (const float*)d_in[7];
  float*            out      = (float*)d_out;

  const int       N = in_sizes[0] / IN_DIM;
  const long long E = in_sizes[1] / 2;

  // workspace carve-up
  char* w = (char*)d_ws;
  float*    h      = (float*)w;                 w += (size_t)N * IN_DIM * 4;
  float*    a_src  = (float*)w;                 w += (size_t)N * HEADS * 4;
  float*    a_dst  = (float*)w;                 w += (size_t)N * HEADS * 4;
  unsigned* amax   = (unsigned*)w;              w += (size_t)N * HEADS * 4;
  float*    denom  = (float*)w;                 w += (size_t)N * HEADS * 4;
  float*    exbuf  = (float*)w;                 w += (size_t)E * HEADS * 4;

  // 1) zero accumulators (harness does not re-poison between replays)
  {
    long long nOut = (long long)N * IN_DIM;
    long long nNH  = (long long)N * HEADS;
    fill_zero_u32<<<dim3((unsigned)((nOut + 255) / 256)), 256, 0, stream>>>((unsigned*)out, nOut);
    fill_zero_u32<<<dim3((unsigned)((nNH  + 255) / 256)), 256, 0, stream>>>(amax, nNH);
    fill_zero_u32<<<dim3((unsigned)((nNH  + 255) / 256)), 256, 0, stream>>>((unsigned*)denom, nNH);
  }

  // 2) h = x @ W  (WMMA fp32)
  {
    dim3 grid((N + 63) / 64, IN_DIM / 64);
    gemm_xw_wmma<<<grid, 128, 0, stream>>>(x, W, h, N);
  }

  // 3) per-node logits
  {
    int total = N * HEADS;
    node_logits<<<(total + 255) / 256, 256, 0, stream>>>(h, att_src, att_dst, a_src, a_dst, N);
  }

  // 4) edge alpha + segment max
  {
    long long total = E * HEADS;
    edge_alpha_max<<<(unsigned)((total + 255) / 256), 256, 0, stream>>>(ei, a_src, a_dst, exbuf, amax, E);
  }

  // 5) exp + segment sum
  {
    long long total = E * HEADS;
    edge_exp_sum<<<(unsigned)((total + 255) / 256), 256, 0, stream>>>(ei, exbuf, amax, denom, E);
  }

  // 6) weighted scatter
  {
    long long total = E * (IN_DIM / 4);
    long long blocks = (total + 255) / 256;
    if (blocks > (1LL << 20)) blocks = (1LL << 20);
    edge_scatter<<<(unsigned)blocks, 256, 0, stream>>>(ei, h, exbuf, denom, out, E);
  }

  // 7) epilogue
  epilogue_ln<<<N, 256, 0, stream>>>(x, bias, ln_scale, ln_bias, out, N);
}